// ModuleFunction_32023276159496
// MI455X (gfx1250) — compile-verified
//
#include <hip/hip_runtime.h>
#include <math.h>

typedef __attribute__((ext_vector_type(2))) float v2f;
typedef __attribute__((ext_vector_type(8))) float v8f;

#define B_ 32
#define L_ 2048
#define H_ 768
#define T_ 12
#define E_ 7
#define EMPTY_IDX_ 6
#define NPAD 32      // padded N (T*2 = 24 used)
#define NUSED 24
#define KPAIRS (H_/2)

// workspace layout in floats
#define VWS_OFF   0
#define VWS_SIZE  (B_*KPAIRS*NPAD*2)        // 786432 floats: packed B matrices
#define ACC_OFF   (VWS_OFF + VWS_SIZE)      // 2 floats: numerator, denominator
#define LOG_OFF   (ACC_OFF + 16)            // 16-float pad keeps logits 16B-aligned
#define LOG_SIZE  (B_*NUSED*L_)             // 1572864 floats

// ---------------------------------------------------------------- init ------
__global__ void k_init(float* __restrict__ ws) {
    int i = blockIdx.x * blockDim.x + threadIdx.x;
    if (i < VWS_SIZE) ws[VWS_OFF + i] = 0.0f;
    if (i < 2)        ws[ACC_OFF + i] = 0.0f;
}

// ------------------------------------------------- phase 1: build v ---------
// one block per (b,t), 256 threads. cW[z] = sum_h ctx[h]*W[e][h][z]
// v[b,t,z,kk] = cW[z]*span_w[e][z][kk] stored as vws[((b*KPAIRS + z/2)*NPAD + 2t+kk)*2 + z%2]
__global__ void k_phase1(const float* __restrict__ seq,
                         const float* __restrict__ Wm,
                         const float* __restrict__ spanw,
                         const int*   __restrict__ turns,
                         const int*   __restrict__ midx,
                         float* __restrict__ vws) {
    __shared__ float ctx[H_];
    const int bt = blockIdx.x;
    const int b  = bt / T_;
    const int t  = bt % T_;
    int sp = turns[bt * 2 + 0];
    sp = min(max(sp, 0), L_ - 1);
    int e = midx[bt];
    e = min(max(e, 0), E_ - 1);

    const float* srow = seq + ((size_t)b * L_ + sp) * H_;
    for (int h = threadIdx.x; h < H_; h += 256) ctx[h] = srow[h];
    __syncthreads();

    const int z0 = threadIdx.x;              // handles z0, z0+256, z0+512
    const float* We = Wm + (size_t)e * H_ * H_;
    float acc0 = 0.f, acc1 = 0.f, acc2 = 0.f;
    for (int h = 0; h < H_; ++h) {
        const float c = ctx[h];
        const float* wr = We + (size_t)h * H_;
        acc0 = fmaf(c, wr[z0      ], acc0);
        acc1 = fmaf(c, wr[z0 + 256], acc1);
        acc2 = fmaf(c, wr[z0 + 512], acc2);
    }
    float accs[3] = {acc0, acc1, acc2};
#pragma unroll
    for (int j = 0; j < 3; ++j) {
        const int z = z0 + j * 256;
        const float s0 = spanw[((size_t)e * H_ + z) * 2 + 0];
        const float s1 = spanw[((size_t)e * H_ + z) * 2 + 1];
        const int kp = z >> 1, lo = z & 1;
        const size_t base = ((size_t)b * KPAIRS + kp) * NPAD;
        vws[(base + 2 * t + 0) * 2 + lo] = accs[j] * s0;
        vws[(base + 2 * t + 1) * 2 + lo] = accs[j] * s1;
    }
}

// ------------------------------------ phase 2: seq @ v via f32 WMMA ---------
// grid (mtile=128, b=32), one wave (32 threads) per block.
// A: 16x4 f32 tile of seq  (lane m = lane&15; VGPRs hold K = 2*half + {0,1})
// B: 4x16 f32 tile of v    (lane n = lane&15; VGPRs hold K = 2*half + {0,1})
// C/D: 16x16 f32           (lane half selects M+8; VGPR v = row offset)
__global__ void __launch_bounds__(32)
k_phase2(const float* __restrict__ seq,
         const float* __restrict__ vws,
         float* __restrict__ logits) {
    const int mtile = blockIdx.x;
    const int b     = blockIdx.y;
    const int lane  = threadIdx.x;
    const int lo    = lane & 15;
    const int half  = lane >> 4;
    const int mbase = mtile * 16;

    const float* arow  = seq + ((size_t)b * L_ + mbase + lo) * H_;
    const float* bbase = vws + (size_t)b * KPAIRS * NPAD * 2;

    v8f c0 = {};
    v8f c1 = {};
    for (int k = 0; k < H_; k += 4) {
        const v2f a = *(const v2f*)(arow + k + 2 * half);
        const int kp = (k >> 1) + half;
        const float* bp = bbase + (size_t)kp * NPAD * 2;
        const v2f b0 = *(const v2f*)(bp + 2 * lo);
        const v2f b1 = *(const v2f*)(bp + 2 * (lo + 16));
        c0 = __builtin_amdgcn_wmma_f32_16x16x4_f32(false, a, false, b0,
                                                   (short)0, c0, false, false);
        c1 = __builtin_amdgcn_wmma_f32_16x16x4_f32(false, a, false, b1,
                                                   (short)0, c1, false, false);
    }

    // each lane holds 8 contiguous L-rows (mbase + 8*half + 0..7) of column n
    const int l0 = mbase + half * 8;
    {
        const int n = lo;                               // 0..15 always used
        float* dst = logits + ((size_t)b * NUSED + n) * L_ + l0;
#pragma unroll
        for (int v = 0; v < 8; ++v) dst[v] = c0[v];
    }
    if (lo < 8) {                                       // columns 16..23
        const int n = lo + 16;
        float* dst = logits + ((size_t)b * NUSED + n) * L_ + l0;
#pragma unroll
        for (int v = 0; v < 8; ++v) dst[v] = c1[v];
    }
}

// ---------------------------- phase 3: logsumexp + CE + masked sum ----------
__global__ void k_phase3(const float* __restrict__ logits,
                         const int*   __restrict__ turns,
                         const int*   __restrict__ kps,
                         const int*   __restrict__ midx,
                         float* __restrict__ acc) {
    __shared__ float red0[256];
    __shared__ float red1[256];
    const int bt  = blockIdx.x;
    const int b   = bt / T_;
    const int t   = bt % T_;
    const int tid = threadIdx.x;
    const float* lg0 = logits + ((size_t)b * NUSED + 2 * t + 0) * L_;
    const float* lg1 = logits + ((size_t)b * NUSED + 2 * t + 1) * L_;

    float m0 = -INFINITY, m1 = -INFINITY;
    for (int l = tid; l < L_; l += 256) {
        m0 = fmaxf(m0, lg0[l]);
        m1 = fmaxf(m1, lg1[l]);
    }
    red0[tid] = m0; red1[tid] = m1;
    __syncthreads();
    for (int s = 128; s > 0; s >>= 1) {
        if (tid < s) {
            red0[tid] = fmaxf(red0[tid], red0[tid + s]);
            red1[tid] = fmaxf(red1[tid], red1[tid + s]);
        }
        __syncthreads();
    }
    const float M0 = red0[0], M1 = red1[0];
    __syncthreads();

    float s0 = 0.f, s1 = 0.f;
    for (int l = tid; l < L_; l += 256) {
        s0 += expf(lg0[l] - M0);
        s1 += expf(lg1[l] - M1);
    }
    red0[tid] = s0; red1[tid] = s1;
    __syncthreads();
    for (int s = 128; s > 0; s >>= 1) {
        if (tid < s) {
            red0[tid] += red0[tid + s];
            red1[tid] += red1[tid + s];
        }
        __syncthreads();
    }

    if (tid == 0) {
        const float lse0 = M0 + logf(red0[0]);
        const float lse1 = M1 + logf(red1[0]);
        const int tgt0 = kps[bt * 2 + 0];
        const int tgt1 = kps[bt * 2 + 1];
        const int sa0 = min(max(tgt0, 0), L_ - 1);
        const int sa1 = min(max(tgt1, 0), L_ - 1);
        const float ce0 = (tgt0 >= 0) ? (lse0 - lg0[sa0]) : 0.f;
        const float ce1 = (tgt1 >= 0) ? (lse1 - lg1[sa1]) : 0.f;
        const float modm = (midx[bt] != EMPTY_IDX_) ? 1.f : 0.f;
        const float tm   = (turns[bt * 2 + 0] >= 0) ? 1.f : 0.f;
        atomicAdd(&acc[0], 0.5f * (ce0 + ce1) * modm * tm);
        atomicAdd(&acc[1], tm);
    }
}

// ---------------------------------------------------------------- final -----
__global__ void k_final(const float* __restrict__ acc, float* __restrict__ out) {
    out[0] = acc[0] / acc[1];
}

// ---------------------------------------------------------------------------
extern "C" void kernel_launch(void* const* d_in, const int* in_sizes, int n_in,
                              void* d_out, int out_size, void* d_ws, size_t ws_size,
                              hipStream_t stream) {
    const float* seq   = (const float*)d_in[0];   // (32,2048,768)
    const float* Wm    = (const float*)d_in[1];   // (7,768,768)
    const float* spanw = (const float*)d_in[2];   // (7,768,2)
    const int*   turns = (const int*)d_in[3];     // (32,12,2)
    const int*   kps   = (const int*)d_in[4];     // (32,12,2)
    const int*   midx  = (const int*)d_in[5];     // (32,12)

    float* ws     = (float*)d_ws;
    float* vws    = ws + VWS_OFF;
    float* acc    = ws + ACC_OFF;
    float* logits = ws + LOG_OFF;
    float* out    = (float*)d_out;

    // zero packed-B (incl. pad columns) + accumulators
    k_init<<<(VWS_SIZE + 255) / 256, 256, 0, stream>>>(ws);
    // build per-(b,t) span vectors
    k_phase1<<<B_ * T_, 256, 0, stream>>>(seq, Wm, spanw, turns, midx, vws);
    // big memory-bound GEMM via f32 WMMA
    k_phase2<<<dim3(L_ / 16, B_), 32, 0, stream>>>(seq, vws, logits);
    // logsumexp + cross-entropy + masking
    k_phase3<<<B_ * T_, 256, 0, stream>>>(logits, turns, kps, midx, acc);
    // scalar loss
    k_final<<<1, 1, 0, stream>>>(acc, out);
}